// EEGGraphConvNet_51273319579928
// MI455X (gfx1250) — compile-verified
//
#include <hip/hip_runtime.h>

// ---------------------------------------------------------------------------
// EEGGraphConvNet forward on MI455X (gfx1250, wave32).
//
// Roofline: ~2.3 GB of traffic dominates (~100us @ 23.3 TB/s); all hot
// intermediates (h1/agg1 32MB, h2/agg2 20MB) fit in the 192MB L2, so the
// random-index edge gathers + ~210M f32 scatter atomics mostly hit L2.
// Dense per-node GEMMs use V_WMMA_F32_16X16X4_F32 chains (exact f32):
//   GEMM1 [N,6]x[6,32]  : per 16-node tile, 2 K-steps x 2 col-tiles = 4 WMMA
//   GEMM2 [N,32]x[32,20]: per 16-node tile, 8 K-steps x 2 col-tiles = 16 WMMA
//     (leaky(agg1+b1) fused into the A-fragment loads -> saves a full pass)
// ---------------------------------------------------------------------------

#define NODES  253952
#define EDGES  (NODES * 16)
#define GRAPHS 4096
#define SLOPE  0.01f
#define BN_EPS 1e-5f

typedef __attribute__((ext_vector_type(2))) float v2f;
typedef __attribute__((ext_vector_type(8))) float v8f;

__device__ __forceinline__ float leaky(float x) { return x >= 0.0f ? x : SLOPE * x; }

// ---------------------------------------------------------------------------
__global__ void zero_f32(float* __restrict__ p, int n) {
    int i = blockIdx.x * blockDim.x + threadIdx.x;
    if (i < n) p[i] = 0.0f;
}

// Zero-pad weights: W1 [6,32] -> W1p [8,32]; W2 [32,20] -> W2p [32,32].
__global__ void pack_weights(const float* __restrict__ W1, const float* __restrict__ W2,
                             float* __restrict__ W1p, float* __restrict__ W2p) {
    int t = threadIdx.x;
    {   // 256 entries of W1p, one per thread
        int k = t >> 5, c = t & 31;
        W1p[t] = (k < 6) ? W1[k * 32 + c] : 0.0f;
    }
    for (int i = t; i < 1024; i += blockDim.x) {
        int k = i >> 5, c = i & 31;
        W2p[i] = (c < 20) ? W2[k * 20 + c] : 0.0f;
    }
}

// ---------------------------------------------------------------------------
// GEMM1: h1[N,32] = x[N,6] @ W1p[8,32]   (K padded 6->8, two 16x16x4 WMMAs/ct)
// One wave per 16-node tile. A layout (32-bit 16x4): lanes 0-15 carry {K0,K1},
// lanes 16-31 carry {K2,K3} of the same M rows. D layout: VGPR r = rows r,r+8.
__global__ void gemm1_wmma(const float* __restrict__ x, const float* __restrict__ W1p,
                           float* __restrict__ h1) {
    int wave  = (blockIdx.x * blockDim.x + threadIdx.x) >> 5;
    int lane  = threadIdx.x & 31;
    int m     = (wave << 4) + (lane & 15);
    int koff  = (lane >= 16) ? 2 : 0;

    const float* xr = x + (size_t)m * 6;
    v2f a0, a1;
    a0.x = xr[koff];
    a0.y = xr[koff + 1];
    int k1 = 4 + koff;                       // K-step 1 covers k=4..7 (6,7 pad)
    a1.x = (k1 < 6) ? xr[k1] : 0.0f;
    a1.y = (k1 + 1 < 6) ? xr[k1 + 1] : 0.0f;

    int n16   = lane & 15;
    int mbase = (wave << 4) + ((lane >> 4) << 3);

    #pragma unroll
    for (int ct = 0; ct < 2; ++ct) {
        int col = ct * 16 + n16;
        v2f b0, b1;
        b0.x = W1p[(koff + 0) * 32 + col];
        b0.y = W1p[(koff + 1) * 32 + col];
        b1.x = W1p[(4 + koff) * 32 + col];
        b1.y = W1p[(5 + koff) * 32 + col];
        v8f c = {};
        c = __builtin_amdgcn_wmma_f32_16x16x4_f32(false, a0, false, b0, (short)0, c, false, false);
        c = __builtin_amdgcn_wmma_f32_16x16x4_f32(false, a1, false, b1, (short)0, c, false, false);
        #pragma unroll
        for (int r = 0; r < 8; ++r)
            h1[(size_t)(mbase + r) * 32 + col] = c[r];
    }
}

// ---------------------------------------------------------------------------
// GEMM2: h2[N,20] = leaky(agg1 + b1)[N,32] @ W2p[32,32-pad]  (8 K-steps)
__global__ void gemm2_wmma(const float* __restrict__ agg1, const float* __restrict__ b1,
                           const float* __restrict__ W2p, float* __restrict__ h2) {
    int wave = (blockIdx.x * blockDim.x + threadIdx.x) >> 5;
    int lane = threadIdx.x & 31;
    int m    = (wave << 4) + (lane & 15);
    int koff = (lane >= 16) ? 2 : 0;

    const float* ar = agg1 + (size_t)m * 32;
    v2f a[8];
    #pragma unroll
    for (int s = 0; s < 8; ++s) {            // fused bias + leaky from layer 1
        int k  = 4 * s + koff;
        a[s].x = leaky(ar[k]     + b1[k]);
        a[s].y = leaky(ar[k + 1] + b1[k + 1]);
    }

    int n16   = lane & 15;
    int mbase = (wave << 4) + ((lane >> 4) << 3);

    #pragma unroll
    for (int ct = 0; ct < 2; ++ct) {
        int col = ct * 16 + n16;
        v8f c = {};
        #pragma unroll
        for (int s = 0; s < 8; ++s) {
            int k = 4 * s + koff;
            v2f b;
            b.x = W2p[k * 32 + col];
            b.y = W2p[(k + 1) * 32 + col];
            c = __builtin_amdgcn_wmma_f32_16x16x4_f32(false, a[s], false, b, (short)0, c, false, false);
        }
        if (col < 20) {
            #pragma unroll
            for (int r = 0; r < 8; ++r)
                h2[(size_t)(mbase + r) * 20 + col] = c[r];
        }
    }
}

// ---------------------------------------------------------------------------
// Edge scatter: agg[dst, :] += w_e * h[src, :], PARTS threads/edge, 4 ch each.
template <int PARTS, int STRIDE>
__global__ void edge_pass(const int* __restrict__ ei, const float* __restrict__ ew,
                          const float* __restrict__ h, float* __restrict__ agg) {
    int tid = blockIdx.x * blockDim.x + threadIdx.x;
    if (tid >= EDGES * PARTS) return;
    int e    = tid / PARTS;
    int part = tid - e * PARTS;
    int s    = ei[e];
    int d    = ei[EDGES + e];
    float w  = ew[e];
    const float4 v = *(const float4*)(h + (size_t)s * STRIDE + part * 4);
    float* ap = agg + (size_t)d * STRIDE + part * 4;
    atomicAdd(ap + 0, w * v.x);
    atomicAdd(ap + 1, w * v.y);
    atomicAdd(ap + 2, w * v.z);
    atomicAdd(ap + 3, w * v.w);
}

// ---------------------------------------------------------------------------
// BN statistics: per-channel sum / sumsq of (agg2 + b2) over all nodes.
__global__ void bn_stats(const float* __restrict__ agg2, const float* __restrict__ b2,
                         float* __restrict__ sums) {  // [0..19]=sum, [20..39]=sumsq
    int n = blockIdx.x * blockDim.x + threadIdx.x;
    float v[20];
    if (n < NODES) {
        const float* r = agg2 + (size_t)n * 20;
        #pragma unroll
        for (int c = 0; c < 20; ++c) v[c] = r[c] + b2[c];
    } else {
        #pragma unroll
        for (int c = 0; c < 20; ++c) v[c] = 0.0f;
    }
    #pragma unroll
    for (int c = 0; c < 20; ++c) {
        float s = v[c], q = v[c] * v[c];
        for (int o = 16; o > 0; o >>= 1) {   // wave32 tree reduce
            s += __shfl_down(s, o, 32);
            q += __shfl_down(q, o, 32);
        }
        if ((threadIdx.x & 31) == 0) {
            atomicAdd(&sums[c],      s);
            atomicAdd(&sums[20 + c], q);
        }
    }
}

__global__ void bn_finalize(const float* __restrict__ sums, const float* __restrict__ gamma,
                            const float* __restrict__ beta, float* __restrict__ scale,
                            float* __restrict__ shift) {
    int c = threadIdx.x;
    if (c < 20) {
        float mean = sums[c] * (1.0f / NODES);
        float var  = sums[20 + c] * (1.0f / NODES) - mean * mean;  // biased var
        float sc   = gamma[c] * rsqrtf(var + BN_EPS);
        scale[c]   = sc;
        shift[c]   = beta[c] - mean * sc;
    }
}

// Normalize + leaky + global_add_pool (batch is sorted; atomics hit L2).
__global__ void bn_pool(const float* __restrict__ agg2, const float* __restrict__ b2,
                        const float* __restrict__ scale, const float* __restrict__ shift,
                        const int* __restrict__ batch, float* __restrict__ pooled) {
    int n = blockIdx.x * blockDim.x + threadIdx.x;
    if (n >= NODES) return;
    int g = batch[n];
    const float* r = agg2 + (size_t)n * 20;
    float* pg = pooled + (size_t)g * 20;
    #pragma unroll
    for (int c = 0; c < 20; ++c) {
        float hn = leaky((r[c] + b2[c]) * scale[c] + shift[c]);
        atomicAdd(&pg[c], hn);
    }
}

// MLP head: [G,20] -> leaky -> [G,10] -> [G,2]. One thread per graph.
__global__ void head(const float* __restrict__ pooled,
                     const float* __restrict__ fc1_w, const float* __restrict__ fc1_b,
                     const float* __restrict__ fc2_w, const float* __restrict__ fc2_b,
                     float* __restrict__ out) {
    int g = blockIdx.x * blockDim.x + threadIdx.x;
    if (g >= GRAPHS) return;
    float p[20];
    #pragma unroll
    for (int k = 0; k < 20; ++k) p[k] = pooled[(size_t)g * 20 + k];
    float hdn[10];
    #pragma unroll
    for (int j = 0; j < 10; ++j) {
        float s = fc1_b[j];
        #pragma unroll
        for (int k = 0; k < 20; ++k) s += p[k] * fc1_w[k * 10 + j];
        hdn[j] = leaky(s);
    }
    #pragma unroll
    for (int o = 0; o < 2; ++o) {
        float s = fc2_b[o];
        #pragma unroll
        for (int j = 0; j < 10; ++j) s += hdn[j] * fc2_w[j * 2 + o];
        out[(size_t)g * 2 + o] = s;
    }
}

// ---------------------------------------------------------------------------
extern "C" void kernel_launch(void* const* d_in, const int* in_sizes, int n_in,
                              void* d_out, int out_size, void* d_ws, size_t ws_size,
                              hipStream_t stream) {
    const float* x     = (const float*)d_in[0];
    const int*   ei    = (const int*)d_in[1];   // [2,E]
    const float* ew    = (const float*)d_in[2];
    const int*   batch = (const int*)d_in[3];
    const float* W1    = (const float*)d_in[4];
    const float* b1    = (const float*)d_in[5];
    const float* W2    = (const float*)d_in[6];
    const float* b2    = (const float*)d_in[7];
    const float* gamma = (const float*)d_in[8];
    const float* beta  = (const float*)d_in[9];
    const float* fc1_w = (const float*)d_in[10];
    const float* fc1_b = (const float*)d_in[11];
    const float* fc2_w = (const float*)d_in[12];
    const float* fc2_b = (const float*)d_in[13];

    // Workspace layout (floats). Zeroed region is contiguous: agg1|agg2|pooled|sums.
    float* ws     = (float*)d_ws;
    float* h1     = ws;                          // N*32
    float* h2     = h1 + (size_t)NODES * 32;     // N*20
    float* W1p    = h2 + (size_t)NODES * 20;     // 256
    float* W2p    = W1p + 256;                   // 1024
    float* agg1   = W2p + 1024;                  // N*32
    float* agg2   = agg1 + (size_t)NODES * 32;   // N*20
    float* pooled = agg2 + (size_t)NODES * 20;   // G*20
    float* sums   = pooled + (size_t)GRAPHS * 20;// 40
    float* scale  = sums + 40;                   // 20
    float* shift  = scale + 20;                  // 20

    const int zcount = NODES * 52 + GRAPHS * 20 + 40;

    zero_f32<<<(zcount + 255) / 256, 256, 0, stream>>>(agg1, zcount);
    pack_weights<<<1, 256, 0, stream>>>(W1, W2, W1p, W2p);

    // 16-node tiles, 8 waves/block -> exact grids (EXEC all-ones for WMMA).
    gemm1_wmma<<<NODES / 16 / 8, 256, 0, stream>>>(x, W1p, h1);
    edge_pass<8, 32><<<(EDGES * 8 + 255) / 256, 256, 0, stream>>>(ei, ew, h1, agg1);
    gemm2_wmma<<<NODES / 16 / 8, 256, 0, stream>>>(agg1, b1, W2p, h2);
    edge_pass<5, 20><<<(EDGES * 5 + 255) / 256, 256, 0, stream>>>(ei, ew, h2, agg2);

    bn_stats<<<NODES / 256, 256, 0, stream>>>(agg2, b2, sums);
    bn_finalize<<<1, 32, 0, stream>>>(sums, gamma, beta, scale, shift);
    bn_pool<<<NODES / 256, 256, 0, stream>>>(agg2, b2, scale, shift, batch, pooled);
    head<<<GRAPHS / 256, 256, 0, stream>>>(pooled, fc1_w, fc1_b, fc2_w, fc2_b, (float*)d_out);
}